// InterpretableMultiHeadAttention_2121713844709
// MI455X (gfx1250) — compile-verified
//
#include <hip/hip_runtime.h>
#include <stdint.h>

// ---------------------------------------------------------------------------
// InterpretableMultiHeadAttention for MI455X (gfx1250), wave32 + WMMA f32
// + async global->LDS double-buffered operand staging (CDNA5 ASYNCcnt path).
//   B=4, S=1024, D=1024, H=16, HS=64
// Memory-bound (~120MB @ 23.3TB/s); f32 WMMA keeps reference precision free.
// ---------------------------------------------------------------------------

constexpr int CB  = 4;
constexpr int CS  = 1024;
constexpr int CD  = 1024;
constexpr int CH  = 16;
constexpr int CHS = 64;

typedef __attribute__((ext_vector_type(2))) float v2f;
typedef __attribute__((ext_vector_type(8))) float v8f;

// One K=4 step of D = A(16x4) * B(4x16) + C  (f32 WMMA, wave32)
__device__ __forceinline__ v8f wmma4(v2f a, v2f b, v8f c) {
  return __builtin_amdgcn_wmma_f32_16x16x4_f32(
      /*neg_a=*/false, a, /*neg_b=*/false, b,
      /*c_mod=*/(short)0, c, /*reuse_a=*/false, /*reuse_b=*/false);
}

// Generic (flat) pointer to LDS byte offset: LDS aperture keeps the offset in
// the low 32 bits (ISA 10.2 aperture mapping).
__device__ __forceinline__ uint32_t lds_off(const void* p) {
  return (uint32_t)(uintptr_t)p;
}

// CDNA5 async copy: 16 bytes per lane, global -> LDS, tracked by ASYNCcnt.
__device__ __forceinline__ void async_ld16(uint32_t lds, const void* g) {
  asm volatile("global_load_async_to_lds_b128 %0, %1, off"
               :: "v"(lds), "v"(g) : "memory");
}
__device__ __forceinline__ void async_wait_le2() {  // prev stage (2 ops/lane) done
  asm volatile("s_wait_asynccnt 0x2" ::: "memory");
}
__device__ __forceinline__ void async_wait_0() {    // all async complete
  asm volatile("s_wait_asynccnt 0x0" ::: "memory");
}

// A-fragment from row-major X[ld] at tile (r0, k0).
// ISA 7.12.2: lane<16 holds row=lane, K={k0,k0+1}; lane>=16 holds K={k0+2,k0+3}
__device__ __forceinline__ v2f ldA(const float* __restrict__ X, int ld,
                                   int r0, int k0, int l15, int lhi) {
  const float* p = X + (size_t)(r0 + l15) * ld + k0 + 2 * lhi;
  v2f a; a.x = p[0]; a.y = p[1];
  return a;
}

// B-fragment from row-major Bm[K][ld] at tile (k0, n0)
__device__ __forceinline__ v2f ldB(const float* __restrict__ Bm, int ld,
                                   int k0, int n0, int l15, int lhi) {
  const float* p = Bm + (size_t)(k0 + 2 * lhi) * ld + n0 + l15;
  v2f b; b.x = p[0]; b.y = p[ld];
  return b;
}

// ---------------------------------------------------------------------------
// Kernel 1: projections.  out[b,h,s,e] = X[b,s,:] @ W[h,:,e] + bias[h,e]
// One wave per 16x16 output tile; streaming GEMM, latency hidden by occupancy.
// Used for q (Hn=16), k (Hn=16), v (Hn=1).
// ---------------------------------------------------------------------------
__global__ __launch_bounds__(256) void proj_kernel(
    const float* __restrict__ X,     // [CB, CS, CD]
    const float* __restrict__ W,     // [Hn, CD, CHS]
    const float* __restrict__ bias,  // [Hn, CHS]
    float* __restrict__ out,         // [CB, Hn, CS, CHS]
    int Hn) {
  const int wave  = blockIdx.x * (blockDim.x >> 5) + (threadIdx.x >> 5);
  const int ntn   = CHS / 16;
  const int ntm   = CS / 16;
  const int total = CB * Hn * ntm * ntn;
  if (wave >= total) return;                  // wave-uniform: EXEC stays full

  int t  = wave;
  const int nt = t % ntn; t /= ntn;
  const int mt = t % ntm; t /= ntm;
  const int h  = t % Hn;
  const int b  = t / Hn;

  const int lane = threadIdx.x & 31;
  const int l15  = lane & 15;
  const int lhi  = lane >> 4;
  const int m0   = mt * 16, n0 = nt * 16;

  const float* Xb = X + (size_t)b * CS * CD;
  const float* Wh = W + (size_t)h * CD * CHS;

  v8f c = {};
#pragma unroll 8
  for (int k0 = 0; k0 < CD; k0 += 4) {
    v2f a  = ldA(Xb, CD, m0, k0, l15, lhi);
    v2f bb = ldB(Wh, CHS, k0, n0, l15, lhi);
    c = wmma4(a, bb, c);
  }

  const float bv = bias[h * CHS + n0 + l15];
  float* O = out + ((size_t)b * Hn + h) * CS * CHS;
#pragma unroll
  for (int i = 0; i < 8; ++i) {
    const int row = m0 + lhi * 8 + i;
    O[(size_t)row * CHS + n0 + l15] = c[i] + bv;
  }
}

// ---------------------------------------------------------------------------
// Kernel 2: fused causal attention + head-mean + output projection.
// One 128-thread block (4 waves) owns (batch b, 16 query rows), loops heads.
// K tiles: per-wave async double-buffered LDS staging (no barriers needed).
// V chunks: cooperative async double-buffered staging (barriered).
// ---------------------------------------------------------------------------
#define BQ 16
#define SPAD 4        // +4 floats/row: kills power-of-two bank conflicts,
                      // keeps rows 16B-aligned for b128 async writes
#define KLD (CHS + SPAD)

__global__ __launch_bounds__(128) void attn_kernel(
    const float* __restrict__ q,   // [CB, CH, CS, CHS]
    const float* __restrict__ k,   // [CB, CH, CS, CHS]
    const float* __restrict__ v,   // [CB, CS, CHS]
    const float* __restrict__ Wo,  // [CHS, CD]
    const float* __restrict__ bo,  // [CD]
    float* __restrict__ out,       // [CB, CS, CD]
    float* __restrict__ attnw) {   // [CB, CS, CS]
  __shared__ __attribute__((aligned(16))) float sS[BQ][CS + SPAD]; // 65.8 KB
  __shared__ __attribute__((aligned(16))) float sV[2][64][KLD];    // 34.8 KB
  __shared__ __attribute__((aligned(16))) float sK[4][2][16][KLD]; // 34.8 KB
  __shared__ __attribute__((aligned(16))) float sO[BQ][KLD];       //  4.4 KB
  __shared__ float red[BQ][8];

  const int tid  = threadIdx.x;
  const int wid  = tid >> 5;             // wave 0..3
  const int lane = tid & 31;
  const int l15  = lane & 15;
  const int lhi  = lane >> 4;

  const int mt = blockIdx.x % (CS / BQ);
  const int b  = blockIdx.x / (CS / BQ);
  const int m0 = mt * BQ;

  const float* vb = v + (size_t)b * CS * CHS;
  float* aw = attnw + ((size_t)b * CS + m0) * CS;

  v8f oacc = {};                         // wave wid owns head-out cols [16w,16w+16)

  for (int h = 0; h < CH; ++h) {
    const float* qh = q + ((size_t)b * CH + h) * CS * CHS;
    const float* kh = k + ((size_t)b * CH + h) * CS * CHS;

    // Q fragments for this head: reused by all 64 score column tiles.
    v2f qfrag[16];
#pragma unroll
    for (int kk = 0; kk < 16; ++kk) qfrag[kk] = ldA(qh, CHS, m0, kk * 4, l15, lhi);

    // ---- scores: S = (Q K^T)/8, causal mask, into LDS ----
    // Per-wave async staging of the 16x64 K tile, double-buffered.
    // Stage = 64 x 16B segments -> 2 ops/lane.
#pragma unroll
    for (int s = lane; s < 64; s += 32) {
      const int row = s >> 2, part = s & 3;
      async_ld16(lds_off(&sK[wid][0][row][part * 4]),
                 kh + (size_t)(wid * 16 + row) * CHS + part * 4);
    }
    for (int j = 0; j < 16; ++j) {
      const int buf = j & 1;
      if (j < 15) {
        const int nn0 = (wid + 4 * (j + 1)) * 16;
#pragma unroll
        for (int s = lane; s < 64; s += 32) {
          const int row = s >> 2, part = s & 3;
          async_ld16(lds_off(&sK[wid][buf ^ 1][row][part * 4]),
                     kh + (size_t)(nn0 + row) * CHS + part * 4);
        }
        async_wait_le2();     // stage j landed (wave-local: no barrier)
      } else {
        async_wait_0();
      }
      const int n0 = (wid + 4 * j) * 16;
      v8f c = {};
#pragma unroll
      for (int kk = 0; kk < CHS; kk += 4) {
        v2f bb;               // B = K^T: row of K per output column
        bb.x = sK[wid][buf][l15][kk + 2 * lhi];
        bb.y = sK[wid][buf][l15][kk + 2 * lhi + 1];
        c = wmma4(qfrag[kk >> 2], bb, c);
      }
#pragma unroll
      for (int i = 0; i < 8; ++i) {
        const int row = lhi * 8 + i;
        const int col = n0 + l15;
        float s = c[i] * 0.125f;                  // 1/sqrt(64)
        if (col > m0 + row) s = -1e9f;            // causal (== tril mask input)
        sS[row][col] = s;
      }
    }
    __syncthreads();

    // ---- softmax over 1024 cols; 8 threads/row, 128-col segments ----
    {
      const int row = tid & 15;
      const int seg = tid >> 4;
      const int c0  = seg * 128;
      float mx = -3.0e38f;
      for (int c1 = c0; c1 < c0 + 128; ++c1) mx = fmaxf(mx, sS[row][c1]);
      red[row][seg] = mx;
      __syncthreads();
      float rmx = red[row][0];
#pragma unroll
      for (int s2 = 1; s2 < 8; ++s2) rmx = fmaxf(rmx, red[row][s2]);
      float sum = 0.f;
      for (int c1 = c0; c1 < c0 + 128; ++c1) {
        const float e = __expf(sS[row][c1] - rmx);
        sS[row][c1] = e;
        sum += e;
      }
      __syncthreads();
      red[row][seg] = sum;
      __syncthreads();
      float rsum = 0.f;
#pragma unroll
      for (int s2 = 0; s2 < 8; ++s2) rsum += red[row][s2];
      const float inv = 1.0f / rsum;
      for (int c1 = c0; c1 < c0 + 128; ++c1) {
        const float p = sS[row][c1] * inv;
        sS[row][c1] = p;
        // head-mean attention weights (block owns this row slab: race-free)
        float* awp = &aw[(size_t)row * CS + c1];
        const float prev = (h == 0) ? 0.f : *awp;
        *awp = prev + p * (1.0f / CH);
      }
      __syncthreads();
    }

    // ---- head_out: oacc += P(16x1024) @ V(1024x64), wave -> 16 cols ----
    // Cooperative async staging of 64-row V chunks, double-buffered.
    // Stage = 256 x 16B segments / 128 threads -> 2 ops/thread.
#pragma unroll
    for (int s = tid; s < 256; s += 128) {
      const int row = s >> 2, part = s & 3;
      async_ld16(lds_off(&sV[0][row][part * 4]),
                 vb + (size_t)row * CHS + part * 4);
    }
    for (int ch = 0; ch < 16; ++ch) {
      const int buf = ch & 1;
      if (ch < 15) {
        const int r0 = (ch + 1) * 64;
#pragma unroll
        for (int s = tid; s < 256; s += 128) {
          const int row = s >> 2, part = s & 3;
          async_ld16(lds_off(&sV[buf ^ 1][row][part * 4]),
                     vb + (size_t)(r0 + row) * CHS + part * 4);
        }
        async_wait_le2();
      } else {
        async_wait_0();
      }
      __syncthreads();        // chunk ch visible to all waves
      const int c0 = ch * 64;
      const int n0 = wid * 16;
#pragma unroll
      for (int kk = 0; kk < 64; kk += 4) {
        v2f a;
        a.x = sS[l15][c0 + kk + 2 * lhi];
        a.y = sS[l15][c0 + kk + 2 * lhi + 1];
        v2f bb;
        bb.x = sV[buf][kk + 2 * lhi][n0 + l15];
        bb.y = sV[buf][kk + 2 * lhi + 1][n0 + l15];
        oacc = wmma4(a, bb, oacc);
      }
      __syncthreads();        // reads done before this buffer is re-staged
    }
  }

  // ---- mean over heads, stage to LDS for the output projection ----
#pragma unroll
  for (int i = 0; i < 8; ++i) sO[lhi * 8 + i][wid * 16 + l15] = oacc[i] * (1.0f / CH);
  __syncthreads();

  // ---- out tile: (16 x 1024) = sO(16x64) @ Wo(64x1024) + bo ----
  for (int j = 0; j < 16; ++j) {
    const int n0 = (wid + 4 * j) * 16;
    v8f c = {};
#pragma unroll
    for (int k0 = 0; k0 < CHS; k0 += 4) {
      v2f a; a.x = sO[l15][k0 + 2 * lhi]; a.y = sO[l15][k0 + 2 * lhi + 1];
      v2f bb = ldB(Wo, CD, k0, n0, l15, lhi);
      c = wmma4(a, bb, c);
    }
    const float bias = bo[n0 + l15];
#pragma unroll
    for (int i = 0; i < 8; ++i) {
      const int row = m0 + lhi * 8 + i;
      out[((size_t)b * CS + row) * CD + n0 + l15] = c[i] + bias;
    }
  }
}

// ---------------------------------------------------------------------------
extern "C" void kernel_launch(void* const* d_in, const int* in_sizes, int n_in,
                              void* d_out, int out_size, void* d_ws, size_t ws_size,
                              hipStream_t stream) {
  (void)in_sizes; (void)n_in; (void)out_size; (void)ws_size;

  const float* queries = (const float*)d_in[0];
  const float* keys    = (const float*)d_in[1];
  const float* values  = (const float*)d_in[2];
  // d_in[3] = mask (tril): applied analytically in-kernel
  const float* Wq = (const float*)d_in[4];
  const float* bq = (const float*)d_in[5];
  const float* Wk = (const float*)d_in[6];
  const float* bk = (const float*)d_in[7];
  const float* Wv = (const float*)d_in[8];
  const float* bv = (const float*)d_in[9];
  const float* Wo = (const float*)d_in[10];
  const float* bo = (const float*)d_in[11];

  float* out   = (float*)d_out;                       // [CB,CS,CD]
  float* attnw = out + (size_t)CB * CS * CD;          // [CB,CS,CS]

  // workspace: q | k | v  (f32)  ~34.6 MB
  float* qp = (float*)d_ws;
  float* kp = qp + (size_t)CB * CH * CS * CHS;
  float* vp = kp + (size_t)CB * CH * CS * CHS;

  const int tiles_qk = CB * CH * (CS / 16) * (CHS / 16);  // 16384
  const int tiles_v  = CB * 1  * (CS / 16) * (CHS / 16);  // 1024
  proj_kernel<<<tiles_qk / 8, 256, 0, stream>>>(queries, Wq, bq, qp, CH);
  proj_kernel<<<tiles_qk / 8, 256, 0, stream>>>(keys,    Wk, bk, kp, CH);
  proj_kernel<<<tiles_v  / 8, 256, 0, stream>>>(values,  Wv, bv, vp, 1);

  attn_kernel<<<CB * (CS / BQ), 128, 0, stream>>>(qp, kp, vp, Wo, bo, out, attnw);
}